// Normalize3D_50946902065846
// MI455X (gfx1250) — compile-verified
//
#include <hip/hip_runtime.h>
#include <hip/hip_bf16.h>
#include <stdint.h>

typedef __attribute__((ext_vector_type(4))) float v4f;

// Per-row math: a = c0..c3, b = c4..c7. Writes 8 floats (two NT b128) to outp.
__device__ __forceinline__ void row_compute_store(v4f a, v4f b, float* __restrict__ outp) {
    float sA = a.x + a.y + a.z + a.w;
    float sB = b.x + b.y + b.z + b.w;
    float gx = 0.25f * (sA - sB);
    float gy = 0.25f * ((a.x + a.y - a.z - a.w) + (b.x + b.y - b.z - b.w));
    float gz = 0.25f * ((a.x - a.y + a.z - a.w) + (b.x - b.y + b.z - b.w));
    float inv = 1.0f / (sqrtf(gx * gx + gy * gy + gz * gz) + 1e-8f);
    v4f na = a * inv;
    v4f nb = b * inv;
    float mn = fminf(fminf(fminf(na.x, na.y), fminf(na.z, na.w)),
                     fminf(fminf(nb.x, nb.y), fminf(nb.z, nb.w)));
    float mx = fmaxf(fmaxf(fmaxf(na.x, na.y), fmaxf(na.z, na.w)),
                     fmaxf(fmaxf(nb.x, nb.y), fmaxf(nb.z, nb.w)));
    float shift = (mn > 1.0f ? mn - 1.0f : 0.0f) + (mx < -1.0f ? mx + 1.0f : 0.0f);
    v4f o0 = na - shift;
    v4f o1 = nb - shift;
    v4f* op = (v4f*)outp;
    __builtin_nontemporal_store(o0, op);
    __builtin_nontemporal_store(o1, op + 1);
}

__global__ __launch_bounds__(256) void Normalize3D_kernel(const float* __restrict__ x,
                                                          float* __restrict__ out,
                                                          int nrows) {
    __shared__ char smem[256 * 32];  // 8 KB: 32 B/row * 256 rows

    const int tid       = threadIdx.x;
    const int wave      = tid >> 5;
    const int lane      = tid & 31;
    const int blockRow0 = (int)(blockIdx.x << 8);   // 256 rows per block
    const int waveRow0  = blockRow0 + (wave << 5);  // 32 rows per wave
    const int row       = waveRow0 + lane;

    const bool fullBlock = (blockRow0 + 256) <= nrows;

    if (fullBlock) {
        // --- CDNA5 async DMA stage: wave's 32 rows (1 KB contiguous) -> LDS ---
        // Lane l fetches flat float4 (f4base+l) and (f4base+32+l): two fully
        // coalesced 512 B wave requests, written to LDS in the same flat order.
        const uint32_t smemBase = (uint32_t)(uintptr_t)(&smem[0]); // low 32b == LDS byte addr
        const uint32_t waveLds  = smemBase + (uint32_t)(wave << 10);
        const size_t   f4base   = (size_t)waveRow0 * 2;

        const v4f* g0 = (const v4f*)x + (f4base + (size_t)lane);
        const v4f* g1 = g0 + 32;
        uint32_t   l0 = waveLds + (uint32_t)(lane << 4);
        uint32_t   l1 = l0 + 512;

        asm volatile("global_load_async_to_lds_b128 %0, %1, off"
                     :: "v"(l0), "v"(g0) : "memory");
        asm volatile("global_load_async_to_lds_b128 %0, %1, off"
                     :: "v"(l1), "v"(g1) : "memory");
        // Producer and consumer are the same wave: ASYNCcnt alone orders LDS
        // visibility, no workgroup barrier required.
        asm volatile("s_wait_asynccnt 0" ::: "memory");

        // --- each lane reads back its own row (ds_load_b128 x2) ---
        const v4f* srow = (const v4f*)(smem + ((size_t)wave << 10) + ((size_t)lane << 5));
        v4f a = srow[0];
        v4f b = srow[1];
        row_compute_store(a, b, out + (size_t)row * 8);
    } else if (row < nrows) {
        // Tail block: coalesced-stage mapping needs full EXEC, fall back to
        // direct per-lane row loads (pairs of b128 still cover whole lines).
        const v4f* rp = (const v4f*)(x + (size_t)row * 8);
        v4f a = rp[0];
        v4f b = rp[1];
        row_compute_store(a, b, out + (size_t)row * 8);
    }
}

extern "C" void kernel_launch(void* const* d_in, const int* in_sizes, int n_in,
                              void* d_out, int out_size, void* d_ws, size_t ws_size,
                              hipStream_t stream) {
    (void)n_in; (void)out_size; (void)d_ws; (void)ws_size;
    const float* x  = (const float*)d_in[0];
    float*       o  = (float*)d_out;
    const int nrows = in_sizes[0] / 8;           // 16,777,216 for the reference shapes
    const int blocks = (nrows + 255) / 256;
    Normalize3D_kernel<<<dim3(blocks), dim3(256), 0, stream>>>(x, o, nrows);
}